// LIFLayer_45363444580786
// MI455X (gfx1250) — compile-verified
//
#include <hip/hip_runtime.h>
#include <math.h>
#include <stdint.h>

// LIF layer scan: I [B,F,L] f32 -> (z, s) each [B,F,L] f32.
// Memory-bound sequential recurrence (hard reset => non-associative, no WMMA).
// CDNA5 path: 6-deep global_load_async_to_lds_b128 pipeline + s_wait_asynccnt,
// LDS-staged coalesced tiles, non-temporal b128 output stores.

namespace {
constexpr int kB = 64;
constexpr int kF = 256;
constexpr int kL = 2048;
constexpr int ROWS = 64;               // rows per block == blockDim.x (2 wave32)
constexpr int CH = 32;                 // time-steps per chunk
constexpr int STR = 36;                // padded LDS row stride (floats); 144 B = 9*16 B
constexpr int NCHUNK = kL / CH;        // 64
constexpr int NBUF = 6;                // async pipeline depth (5 chunks in flight)
constexpr int TILE = ROWS * STR;       // 2304 floats = 9216 B per tile
constexpr int ITERS = (ROWS * (CH / 4)) / ROWS;  // 8 coop float4 segments/thread
constexpr float BETA = 15.0f;
}

typedef float v4f __attribute__((ext_vector_type(4)));

__device__ __forceinline__ uint32_t lds_addr_of(const void* p) {
  // Low 32 bits of the flat shared-aperture address are the LDS byte offset.
  return (uint32_t)(uintptr_t)p;
}

__device__ __forceinline__ void async_tile_load(const float* __restrict__ gbase,
                                                uint32_t ldsbase, int rowbase,
                                                int k, int t) {
#pragma unroll
  for (int i = 0; i < ITERS; ++i) {
    int seg = i * ROWS + t;       // 0..511
    int r   = seg >> 3;           // row within tile (8 float4 segs per row)
    int c4  = seg & 7;            // float4 column within chunk
    const float* g = gbase + (size_t)(rowbase + r) * kL + (size_t)k * CH + c4 * 4;
    uint32_t l = ldsbase + (uint32_t)((r * STR + c4 * 4) * sizeof(float));
    // async DMA: 16 B per lane; LDS dest byte addr in VGPR %0, 64-bit global addr %1
    asm volatile("global_load_async_to_lds_b128 %0, %1, off"
                 :: "v"(l), "v"(g)
                 : "memory");
  }
}

__device__ __forceinline__ void wait_async(int ahead) {
  switch (ahead) {
    case 5:  asm volatile("s_wait_asynccnt 40" ::: "memory"); break;
    case 4:  asm volatile("s_wait_asynccnt 32" ::: "memory"); break;
    case 3:  asm volatile("s_wait_asynccnt 24" ::: "memory"); break;
    case 2:  asm volatile("s_wait_asynccnt 16" ::: "memory"); break;
    case 1:  asm volatile("s_wait_asynccnt 8"  ::: "memory"); break;
    default: asm volatile("s_wait_asynccnt 0"  ::: "memory"); break;
  }
}

__global__ __launch_bounds__(ROWS) void lif_scan_kernel(
    const float* __restrict__ I, const float* __restrict__ raw_tau,
    const float* __restrict__ thrp, float* __restrict__ zout,
    float* __restrict__ sout) {
  __shared__ __align__(16) float sbuf[NBUF][TILE];  // async input ring
  __shared__ __align__(16) float szf[TILE];
  __shared__ __align__(16) float ssf[TILE];

  const int t = threadIdx.x;
  const int rowbase = blockIdx.x * ROWS;
  const int f = (rowbase + t) & (kF - 1);

  // Per-row decay: tau = softplus(raw_tau) + 1e-4 ; alpha = exp(-dt/tau)
  float x = raw_tau[f];
  float sp = fmaxf(x, 0.0f) + log1pf(expf(-fabsf(x)));
  float tau = sp + 1e-4f;
  float alpha = expf(-1.0f / tau);
  float om = 1.0f - alpha;
  float thr = thrp[0];
  float zb = -BETA * thr;   // z = fma(BETA, v_pre, zb)

  uint32_t ldsring[NBUF];
#pragma unroll
  for (int b = 0; b < NBUF; ++b) ldsring[b] = lds_addr_of(&sbuf[b][0]);

  // Prologue: fill the pipeline with chunks 0..NBUF-1
#pragma unroll
  for (int b = 0; b < NBUF; ++b) async_tile_load(I, ldsring[b], rowbase, b, t);

  float v = 0.0f;
  int bi = 0;  // ring index == k % NBUF
  for (int k = 0; k < NCHUNK; ++k) {
    // Wait until chunk k's 8 loads retired; keep up to 5 later chunks in flight.
    int ahead = NCHUNK - 1 - k;
    if (ahead > NBUF - 1) ahead = NBUF - 1;
    wait_async(ahead);
    __syncthreads();   // other wave's loads visible + staging-buffer reuse safety

    const float* inb = &sbuf[bi][0];
#pragma unroll
    for (int c = 0; c < CH; ++c) {
      float it = inb[t * STR + c];
      float vp = fmaf(alpha, v, om * it);
      bool spike = (vp >= thr);
      szf[t * STR + c] = fmaf(BETA, vp, zb);
      ssf[t * STR + c] = spike ? 1.0f : 0.0f;
      v = spike ? 0.0f : vp;
    }
    __syncthreads();   // tile consumed; z/s staging complete

    // Refill the just-consumed ring slot with chunk k+NBUF (fire and forget).
    if (k + NBUF < NCHUNK) {
      async_tile_load(I, ldsring[bi], rowbase, k + NBUF, t);
    }

    // Coalesced non-temporal b128 stores (8 lanes cover one 128 B line).
#pragma unroll
    for (int i = 0; i < ITERS; ++i) {
      int seg = i * ROWS + t;
      int r   = seg >> 3;
      int c4  = seg & 7;
      int lidx = r * STR + c4 * 4;
      size_t goff = (size_t)(rowbase + r) * kL + (size_t)k * CH + c4 * 4;
      v4f zv = *(const v4f*)&szf[lidx];
      v4f sv = *(const v4f*)&ssf[lidx];
      __builtin_nontemporal_store(zv, (v4f*)&zout[goff]);
      __builtin_nontemporal_store(sv, (v4f*)&sout[goff]);
    }

    bi = (bi + 1 == NBUF) ? 0 : bi + 1;
  }
}

extern "C" void kernel_launch(void* const* d_in, const int* in_sizes, int n_in,
                              void* d_out, int out_size, void* d_ws, size_t ws_size,
                              hipStream_t stream) {
  (void)in_sizes; (void)n_in; (void)out_size; (void)d_ws; (void)ws_size;
  const float* I       = (const float*)d_in[0];
  const float* raw_tau = (const float*)d_in[1];
  const float* thr     = (const float*)d_in[2];
  float* z = (float*)d_out;
  float* s = z + (size_t)kB * kF * kL;

  dim3 grid((kB * kF) / ROWS);   // 256 blocks
  dim3 block(ROWS);              // 64 threads (2 wave32)
  hipLaunchKernelGGL(lif_scan_kernel, grid, block, 0, stream, I, raw_tau, thr, z, s);
}